// BodyAgnosticNACPG_26139170963721
// MI455X (gfx1250) — compile-verified
//
#include <hip/hip_runtime.h>
#include <hip/hip_bf16.h>
#include <math.h>

// ---------------------------------------------------------------------------
// BodyAgnosticNACPG — O(N) reformulation of the O(N^2) reference scan.
//
//   R[i,n] = Rot(phase_i - phase_n) = Rot(p_i) * Rot(-p_n)
//   coup_i = COUP/deg_i * Rot(p_i) @ (S - u_i),  u_n = Rot(-p_n) @ xy_n
//   S maintained incrementally across the sequential scan.
//
// Kernel 1 (device-wide): deg_i = max(rowsum(adj & ~eye), 1)  (16 MB of bool)
// Kernel 2 (1 workgroup): parallel precompute into ~180 KB of the 320 KB WGP
//   LDS (async global->LDS staging for amp/b), wave32 tree reduction for the
//   initial S, then a single-lane 4096-step recurrence fed by ds_load_b128
//   with +1-iteration software pipelining.
// ---------------------------------------------------------------------------

#define ALPHA_C 0.45f
#define DT_C    0.01f
#define COUP_C  0.08f
#define DIFF_C  10.0f
#define EPS_C   1e-9f

#if defined(__HIP_DEVICE_COMPILE__) && defined(__gfx1250__)
  #if __has_builtin(__builtin_amdgcn_global_load_async_to_lds_b32)
    #define HAVE_ASYNC_LDS 1
  #endif
  #if __has_builtin(__builtin_amdgcn_s_wait_asynccnt)
    #define HAVE_WAIT_ASYNC 1
  #endif
#endif

typedef __attribute__((address_space(1))) int as1_int;   // "__device__ int"
typedef __attribute__((address_space(3))) int as3_int;   // "__shared__ int"

// CDNA5 async global -> LDS copy (ASYNCcnt-tracked). Generic pointers carry
// the AS1 VA in full 64 bits and the LDS byte offset in the low 32 bits, so
// the via-integer casts are exact.
__device__ __forceinline__ void async_g2l_b32(const float* g, float* l) {
#if defined(HAVE_ASYNC_LDS)
  as1_int* gp = (as1_int*)(unsigned long long)(size_t)g;
  as3_int* lp = (as3_int*)(unsigned int)(size_t)l;
  __builtin_amdgcn_global_load_async_to_lds_b32(gp, lp, 0, 0);
#else
  *l = *g;  // synchronous fallback if the builtin is absent
#endif
}

__device__ __forceinline__ void wait_async_zero() {
#if defined(HAVE_WAIT_ASYNC)
  __builtin_amdgcn_s_wait_asynccnt(0);
#else
  asm volatile("s_wait_asynccnt 0x0" ::: "memory");
#endif
}

// ---------------------------------------------------------------------------
// Kernel 1: one block per adjacency row; deg_i from numpy-bool (1 byte) mask.
// Byte-sum trick: bytes are 0/1, so ((w & 0x01010101) * 0x01010101) >> 24
// sums the 4 lanes of a dword. Integer shared atomic => deterministic.
// ---------------------------------------------------------------------------
__global__ void __launch_bounds__(256) deg_rows_kernel(
    const unsigned char* __restrict__ mask, int n, float* __restrict__ cdeg)
{
  __shared__ int red;
  if (threadIdx.x == 0) red = 0;
  __syncthreads();

  const int row = blockIdx.x;
  const unsigned* m32 = (const unsigned*)(mask + (size_t)row * (size_t)n);
  const int words = n >> 2;
  int s = 0;
  for (int j = threadIdx.x; j < words; j += blockDim.x) {
    unsigned v = m32[j] & 0x01010101u;
    s += (int)((v * 0x01010101u) >> 24);
  }
  atomicAdd(&red, s);
  __syncthreads();

  if (threadIdx.x == 0) {
    int tot = red - (int)(mask[(size_t)row * (size_t)n + row] & 1);  // ~eye
    float deg = (tot < 1) ? 1.0f : (float)tot;
    cdeg[row] = COUP_C / deg;   // store COUP/deg directly
  }
}

// ---------------------------------------------------------------------------
// Kernel 2: precompute + wave32 reduction + single-lane sequential scan.
// Dynamic LDS layout (floats):
//   sA   : (n+1) float4 {cosp, sinp, x0, y0}
//   sB   : (n+1) float4 {a, beta, xdo0, xdo1}
//   sAmp : n   (async-staged)   sBv : n (async-staged)   sCd : n (COUP/deg)
//   sRed : 64  (per-wave partial sums of u)
// ---------------------------------------------------------------------------
__global__ void __launch_bounds__(1024) nacpg_scan_kernel(
    const float* __restrict__ phase, const float* __restrict__ amp,
    const float* __restrict__ w,     const float* __restrict__ ha,
    const float* __restrict__ bvec,  const float* __restrict__ xy,
    const float* __restrict__ xdo,   const float* __restrict__ cdeg,
    const unsigned char* __restrict__ mask, int n, float* __restrict__ out)
{
  extern __shared__ float smem[];
  float4* sA   = (float4*)smem;
  float4* sB   = sA + (n + 1);
  float*  sAmp = (float*)(sB + (n + 1));
  float*  sBv  = sAmp + n;
  float*  sCd  = sBv + n;
  float*  sRed = sCd + n;

  const int tid = threadIdx.x;
  const int nt  = blockDim.x;

  // ---- async staging of the pass-through arrays straight into LDS --------
  for (int i = tid; i < n; i += nt) {
    async_g2l_b32(amp  + i, sAmp + i);
    async_g2l_b32(bvec + i, sBv  + i);
  }

  // ---- parallel precompute + partial sums of u = Rot(-p) @ xy ------------
  float sx = 0.f, sy = 0.f;
  for (int i = tid; i < n; i += nt) {
    float p = phase[i];
    float sp, cp;
    sincosf(p, &sp, &cp);
    float x  = xy[2 * i + 0],  y  = xy[2 * i + 1];
    float d0 = xdo[2 * i + 0], d1 = xdo[2 * i + 1];
    float r2   = x * x + y * y;
    float a    = ALPHA_C * (1.f - r2 * r2);
    float zeta = 1.f - ha[i] * ((d0 + EPS_C) / (fabsf(d0) + EPS_C));
    float bt   = w[i] / (zeta + EPS_C);
    sA[i] = make_float4(cp, sp, x, y);
    sB[i] = make_float4(a, bt, d0, d1);
    sx += cp * x + sp * y;          // u_x
    sy += cp * y - sp * x;          // u_y
  }

  // ---- COUP/deg into LDS (from ws if kernel 1 ran, else compute here) ----
  if (cdeg) {
    for (int i = tid; i < n; i += nt) sCd[i] = cdeg[i];
  } else {
    for (int i = tid; i < n; i += nt) {
      const unsigned* m32 = (const unsigned*)(mask + (size_t)i * (size_t)n);
      int s = 0;
      for (int j = 0; j < (n >> 2); ++j) {
        unsigned v = m32[j] & 0x01010101u;
        s += (int)((v * 0x01010101u) >> 24);
      }
      s -= (int)(mask[(size_t)i * (size_t)n + i] & 1);
      sCd[i] = COUP_C / ((s < 1) ? 1.0f : (float)s);
    }
  }

  // ---- deterministic wave32 xor-tree reduction ---------------------------
  for (int m = 16; m > 0; m >>= 1) {
    sx += __shfl_xor(sx, m, 32);
    sy += __shfl_xor(sy, m, 32);
  }
  if ((tid & 31) == 0) {
    sRed[2 * (tid >> 5) + 0] = sx;
    sRed[2 * (tid >> 5) + 1] = sy;
  }
  if (tid == 0) {  // pad entry for +1 software-pipeline prefetch
    sA[n] = make_float4(0.f, 0.f, 0.f, 0.f);
    sB[n] = make_float4(0.f, 0.f, 0.f, 0.f);
  }

  wait_async_zero();   // drain ASYNCcnt before the barrier publishes LDS
  __syncthreads();
  if (tid != 0) return;

  // ---- ordered final sum of per-wave partials ----------------------------
  float Sx = 0.f, Sy = 0.f;
  const int nwaves = nt >> 5;
  for (int v = 0; v < nwaves; ++v) { Sx += sRed[2 * v]; Sy += sRed[2 * v + 1]; }

  // ---- the sequential recurrence (latency-bound, LDS-fed) ----------------
  float4 A = sA[0], B = sB[0];
  for (int i = 0; i < n; ++i) {
    float4 An = sA[i + 1], Bn = sB[i + 1];   // ds_load_b128 one iter ahead
    float cp = A.x, sp = A.y, x = A.z, y = A.w;
    float a = B.x, bt = B.y, d0 = B.z, d1 = B.w;
    float cd = sCd[i];

    float ux = cp * x + sp * y;              // u_i (pre-update)
    float uy = cp * y - sp * x;
    float vx = Sx - ux, vy = Sy - uy;        // neighbor sum (aw excludes self)
    float cx = cd * (cp * vx - sp * vy);     // Rot(p_i) @ v, scaled
    float cy = cd * (sp * vx + cp * vy);
    float lx = a * x - bt * y;               // K_i @ x_i
    float ly = bt * x + a * y;
    float dx = fminf(fmaxf(lx + cx, d0 - DIFF_C), d0 + DIFF_C);
    float dy = fminf(fmaxf(ly + cy, d1 - DIFF_C), d1 + DIFF_C);
    float xn = x + DT_C * dx;
    float yn = y + DT_C * dy;

    out[i] = sAmp[i] * yn + sBv[i];          // angle_i

    Sx += (cp * xn + sp * yn) - ux;          // S := S - u_old + u_new
    Sy += (cp * yn - sp * xn) - uy;
    A = An; B = Bn;
  }
}

// ---------------------------------------------------------------------------
extern "C" void kernel_launch(void* const* d_in, const int* in_sizes, int n_in,
                              void* d_out, int out_size, void* d_ws, size_t ws_size,
                              hipStream_t stream)
{
  const float* phase = (const float*)d_in[0];
  const float* amp   = (const float*)d_in[1];
  const float* w     = (const float*)d_in[2];
  const float* ha    = (const float*)d_in[3];
  const float* bvec  = (const float*)d_in[4];
  const float* xy    = (const float*)d_in[5];
  const float* xdo   = (const float*)d_in[6];
  const unsigned char* mask = (const unsigned char*)d_in[7];  // numpy bool = 1B
  float* out = (float*)d_out;
  const int n = in_sizes[0];

  float* cdeg = nullptr;
  if (d_ws && ws_size >= (size_t)n * sizeof(float)) {
    cdeg = (float*)d_ws;
    deg_rows_kernel<<<n, 256, 0, stream>>>(mask, n, cdeg);  // 16 MB @ full HBM BW
  }

  const size_t smem_bytes =
      (size_t)(8 * (n + 1) + 3 * n + 64) * sizeof(float);   // ~180.5 KB < 320 KB
  (void)hipFuncSetAttribute((const void*)nacpg_scan_kernel,
                            hipFuncAttributeMaxDynamicSharedMemorySize,
                            (int)smem_bytes);
  nacpg_scan_kernel<<<1, 1024, smem_bytes, stream>>>(
      phase, amp, w, ha, bvec, xy, xdo, cdeg, mask, n, out);
}